// NR_GraphAttentionMu_52853867545025
// MI455X (gfx1250) — compile-verified
//
#include <hip/hip_runtime.h>
#include <hip/hip_bf16.h>
#include <math.h>

// ---------------------------------------------------------------------------
// NR_GraphAttentionMu for MI455X (gfx1250, wave32)
//
// outputs = [tanh(features) | feats1 | feats2], proxy/gate tail is dead code.
// Memory-bound gather/scatter (~1.5KB moved per edge per layer, ~0.4 flop/B),
// both 51MB feature matrices are L2-resident (192MB L2).
//
// Kernels:
//   1. rowptr_kernel      : CSR row ranges via binary search (rows are sorted)
//   2. tanh_feats_kernel  : feats0 = tanh(features) -> out[:,0:128]
//   per layer l:
//   3. edge_att_wmma      : 16-edge tiles; u=|fc[r]+fc[c]| (inf->0) staged in
//                           LDS; V_WMMA_F32_16X16X4_F32 computes diag(U U^T),
//                           diag(U Nh^T), U w  ->  per-edge 1/||u||, dot, att.
//                           All reductions live on the matrix pipe.
//   4. node_agg_kernel    : wave-per-node (deterministic, no float atomics):
//                           segment softmax from stored att, then
//                           acc += e^(att-m) * (nh - 2*dot*ce), tanh(acc/s),
//                           written into out[:,128*(l+1):...] which is also
//                           the next layer's feature storage.
// ---------------------------------------------------------------------------

#define N_NODES 100000
#define N_EDGE  800000
#define NDIM    128
#define FDIM    384   /* NDIM * (DEPTH+1) */
#define EPS     1e-12f

typedef __attribute__((ext_vector_type(2))) float v2f;
typedef __attribute__((ext_vector_type(8))) float v8f;

__device__ __forceinline__ float absz(float x) {
    float a = fabsf(x);
    return isinf(a) ? 0.0f : a;   // reference: where(isinf(ce), 0, ce)
}

// -------------------------------------------------------------- row_ptr ----
__global__ __launch_bounds__(256) void rowptr_kernel(
    const int* __restrict__ row, int* __restrict__ rp, int n_nodes, int n_edges)
{
    int n = blockIdx.x * blockDim.x + threadIdx.x;
    if (n > n_nodes) return;
    int lo = 0, hi = n_edges;                 // lower_bound(row, n)
    while (lo < hi) {
        int mid = (lo + hi) >> 1;
        if (row[mid] < n) lo = mid + 1; else hi = mid;
    }
    rp[n] = lo;
}

// ----------------------------------------------------------- tanh stage ----
__global__ __launch_bounds__(256) void tanh_feats_kernel(
    const float* __restrict__ f, float* __restrict__ out, int total)
{
    int i = blockIdx.x * blockDim.x + threadIdx.x;
    if (i >= total) return;
    int n = i >> 7, d = i & 127;
    out[(size_t)n * FDIM + d] = tanhf(f[i]);
}

// ------------------------------------------------------ edge WMMA stage ----
// One wave = 16 edges. LDS row stride 132 floats: 4-bank shift per row ->
// conflict-free float2 fragment reads; 528B row = 8B aligned for float2.
#define WPB 2   /* waves per block */

__global__ __launch_bounds__(64) void edge_att_wmma_kernel(
    const int* __restrict__ row, const int* __restrict__ col,
    const float* __restrict__ fc, int fcStride,
    const float* __restrict__ ft, int ftStride,
    const float* __restrict__ attnw,
    float* __restrict__ attO, float* __restrict__ dotO, float* __restrict__ ninvO,
    int n_edges)
{
    __shared__ float u [WPB][16][132];
    __shared__ float nh[WPB][16][132];
    __shared__ float wsh[128];
    __shared__ float tmpN[WPB][16], tmpD[WPB][16], tmpA[WPB][16];

    const int tid  = threadIdx.x;
    const int wv   = tid >> 5;
    const int lane = tid & 31;
    const int e_base = (blockIdx.x * WPB + wv) * 16;
    if (e_base >= n_edges) return;            // wave-uniform (tiles exact)

    for (int i = tid; i < 128; i += 64) wsh[i] = attnw[i];

    // Stage u = |fc[row]+fc[col]| (inf->0) and nh = ft[col]; one coalesced
    // 512B float4-per-lane load per row (wave32 covers a full 128-f32 row).
    for (int m = 0; m < 16; ++m) {
        const int e = e_base + m;
        const int r = row[e], c = col[e];
        const float4 fr = *(const float4*)(fc + (size_t)r * fcStride + lane * 4);
        const float4 fg = *(const float4*)(fc + (size_t)c * fcStride + lane * 4);
        const float4 nv = *(const float4*)(ft + (size_t)c * ftStride + lane * 4);
        u [wv][m][lane*4+0] = absz(fr.x + fg.x);
        u [wv][m][lane*4+1] = absz(fr.y + fg.y);
        u [wv][m][lane*4+2] = absz(fr.z + fg.z);
        u [wv][m][lane*4+3] = absz(fr.w + fg.w);
        nh[wv][m][lane*4+0] = nv.x;
        nh[wv][m][lane*4+1] = nv.y;
        nh[wv][m][lane*4+2] = nv.z;
        nh[wv][m][lane*4+3] = nv.w;
    }
    __syncthreads();

    const int mn  = lane & 15;   // A: row m ; B: col n  (layouts are symmetric)
    const int khi = lane >> 4;   // k-pair select

#if __has_builtin(__builtin_amdgcn_wmma_f32_16x16x4_f32)
    v8f cN = {}; v8f cD = {}; v8f cW = {};
    #pragma unroll
    for (int t = 0; t < 32; ++t) {
        const int idx = t * 4 + khi * 2;
        v2f a;  a.x  = u [wv][mn][idx]; a.y  = u [wv][mn][idx + 1];
        v2f bd; bd.x = nh[wv][mn][idx]; bd.y = nh[wv][mn][idx + 1];
        v2f bw; bw.x = wsh[idx];        bw.y = wsh[idx + 1];
        // C[m][n] += sum_k A[m][k]*B[k][n] ; diag gives per-edge reductions.
        cN = __builtin_amdgcn_wmma_f32_16x16x4_f32(false, a, false, a,  (short)0, cN, false, false);
        cD = __builtin_amdgcn_wmma_f32_16x16x4_f32(false, a, false, bd, (short)0, cD, false, false);
        cW = __builtin_amdgcn_wmma_f32_16x16x4_f32(false, a, false, bw, (short)0, cW, false, false);
    }
    // C/D layout: lanes 0-15: vgpr v = C[M=v][N=lane]; lanes 16-31: C[M=8+v][N=lane-16]
    #pragma unroll
    for (int j = 0; j < 8; ++j) {
        if (lane == j)      { tmpN[wv][j]   = cN[j]; tmpD[wv][j]   = cD[j]; }   // diag m=j
        if (lane == 24 + j) { tmpN[wv][8+j] = cN[j]; tmpD[wv][8+j] = cD[j]; }   // diag m=8+j
    }
    if (lane == 0) {
        #pragma unroll
        for (int j = 0; j < 8; ++j) tmpA[wv][j] = cW[j];       // att_0..7
    }
    if (lane == 16) {
        #pragma unroll
        for (int j = 0; j < 8; ++j) tmpA[wv][8 + j] = cW[j];   // att_8..15
    }
#else
    // Scalar fallback: lanes 0..15 each reduce one edge serially from LDS.
    if (lane < 16) {
        float sn = 0.f, sd = 0.f, sw = 0.f;
        for (int k = 0; k < 128; ++k) {
            float uv = u[wv][lane][k];
            sn += uv * uv;
            sd += uv * nh[wv][lane][k];
            sw += uv * wsh[k];
        }
        tmpN[wv][lane] = sn; tmpD[wv][lane] = sd; tmpA[wv][lane] = sw;
    }
#endif
    __syncthreads();

    if (lane < 16) {
        const int e = e_base + lane;
        const float inv = 1.0f / fmaxf(sqrtf(tmpN[wv][lane]), EPS);
        attO[e]  = tmpA[wv][lane] * inv;   // att  = (u.w)/max(||u||,eps)
        dotO[e]  = tmpD[wv][lane] * inv;   // dot  = (u.nh)/max(||u||,eps)
        ninvO[e] = inv;                    // for ce = u * inv
    }
}

// ------------------------------------------------------ node aggregation ---
__global__ __launch_bounds__(256) void node_agg_kernel(
    const int* __restrict__ col, const int* __restrict__ rp,
    const float* __restrict__ fc, int fcStride,
    const float* __restrict__ ft, int ftStride,
    const float* __restrict__ attA, const float* __restrict__ dotA,
    const float* __restrict__ ninvA,
    float* __restrict__ out, int outOff, int n_nodes)
{
    const int wv   = threadIdx.x >> 5;
    const int lane = threadIdx.x & 31;
    const int n    = blockIdx.x * 8 + wv;
    if (n >= n_nodes) return;                     // wave-uniform

    float* dst = out + (size_t)n * FDIM + outOff + lane * 4;
    const int lo = rp[n], hi = rp[n + 1];
    if (lo == hi) {                               // empty segment -> tanh(0)=0
        dst[0] = 0.f; dst[1] = 0.f; dst[2] = 0.f; dst[3] = 0.f;
        return;
    }

    const float4 fcn = *(const float4*)(fc + (size_t)n * fcStride + lane * 4);

    // segment max / sum-exp (deterministic wave reductions)
    float mx = -INFINITY;
    for (int e = lo + lane; e < hi; e += 32) mx = fmaxf(mx, attA[e]);
    for (int o = 16; o; o >>= 1) mx = fmaxf(mx, __shfl_xor(mx, o, 32));
    float s = 0.f;
    for (int e = lo + lane; e < hi; e += 32) s += expf(attA[e] - mx);
    for (int o = 16; o; o >>= 1) s += __shfl_xor(s, o, 32);

    float4 acc = {0.f, 0.f, 0.f, 0.f};
    for (int e = lo; e < hi; ++e) {               // serial -> deterministic
        const float coef = expf(attA[e] - mx);
        const float dt2  = 2.0f * dotA[e];
        const float ninv = ninvA[e];
        const int   c    = col[e];
        const float4 fg  = *(const float4*)(fc + (size_t)c * fcStride + lane * 4);
        const float4 nv  = *(const float4*)(ft + (size_t)c * ftStride + lane * 4);
        acc.x += coef * (nv.x - dt2 * (absz(fcn.x + fg.x) * ninv));
        acc.y += coef * (nv.y - dt2 * (absz(fcn.y + fg.y) * ninv));
        acc.z += coef * (nv.z - dt2 * (absz(fcn.z + fg.z) * ninv));
        acc.w += coef * (nv.w - dt2 * (absz(fcn.w + fg.w) * ninv));
    }
    const float is = 1.0f / s;
    dst[0] = tanhf(acc.x * is);
    dst[1] = tanhf(acc.y * is);
    dst[2] = tanhf(acc.z * is);
    dst[3] = tanhf(acc.w * is);
}

// ------------------------------------------------------------- launcher ----
extern "C" void kernel_launch(void* const* d_in, const int* in_sizes, int n_in,
                              void* d_out, int out_size, void* d_ws, size_t ws_size,
                              hipStream_t stream) {
    const float* features   = (const float*)d_in[0];
    const int*   adj        = (const int*)d_in[2];
    const float* features_c = (const float*)d_in[5];
    const float* attn       = (const float*)d_in[6];   // [2,128]
    float* out = (float*)d_out;

    const int* row = adj;
    const int* col = adj + N_EDGE;

    float* attA  = (float*)d_ws;
    float* dotA  = attA + N_EDGE;
    float* ninvA = dotA + N_EDGE;
    int*   rp    = (int*)(ninvA + N_EDGE);             // N_NODES+1 ints

    rowptr_kernel<<<(N_NODES + 1 + 255) / 256, 256, 0, stream>>>(row, rp, N_NODES, N_EDGE);
    tanh_feats_kernel<<<(N_NODES * NDIM + 255) / 256, 256, 0, stream>>>(
        features, out, N_NODES * NDIM);

    const int eBlocks = N_EDGE / (16 * WPB);           // 800000/32 = 25000 exact
    const int nBlocks = (N_NODES + 7) / 8;

    // layer 0: fc = features_c (stride 128), feats = out[:,0:128] (stride 384)
    edge_att_wmma_kernel<<<eBlocks, 64, 0, stream>>>(
        row, col, features_c, NDIM, out, FDIM, attn,
        attA, dotA, ninvA, N_EDGE);
    node_agg_kernel<<<nBlocks, 256, 0, stream>>>(
        col, rp, features_c, NDIM, out, FDIM,
        attA, dotA, ninvA, out, NDIM, N_NODES);        // writes out[:,128:256]

    // layer 1: fc = feats = out[:,128:256]
    edge_att_wmma_kernel<<<eBlocks, 64, 0, stream>>>(
        row, col, out + NDIM, FDIM, out + NDIM, FDIM, attn + NDIM,
        attA, dotA, ninvA, N_EDGE);
    node_agg_kernel<<<nBlocks, 256, 0, stream>>>(
        col, rp, out + NDIM, FDIM, out + NDIM, FDIM,
        attA, dotA, ninvA, out, 2 * NDIM, N_NODES);    // writes out[:,256:384]
}